// ToeplitzLieAttention_28750511079449
// MI455X (gfx1250) — compile-verified
//
#include <hip/hip_runtime.h>
#include <math.h>

// ToeplitzLieAttention fused kernel for gfx1250 (MI455X), fp32 WMMA path.
// Shapes fixed by the reference: B=2, L=1024, H=8, E=64.
// 1024 blocks x 4 wave32; each block owns a 16-row tile of one (b,h) head,
// the 4 waves split the 64 column tiles (interleaved) and combine via LDS.

typedef __attribute__((ext_vector_type(2))) float v2f;
typedef __attribute__((ext_vector_type(8))) float v8f;

#define B_  2
#define L_  1024
#define H_  8
#define E_  64
#define RS  (H_ * E_)      // row stride in (B,L,H,E) layout = 512 floats
#define TT  (L_ / 16)      // 64 column tiles
#define NW  4              // waves per block

#define LOG2E_ 1.4426950408889634f
#define LN2_   0.6931471805599453f

__device__ __forceinline__ v8f wmma4(v2f a, v2f b, v8f c) {
  // D = A(16x4,f32) * B(4x16,f32) + C(16x16,f32)
  return __builtin_amdgcn_wmma_f32_16x16x4_f32(
      false, a, false, b, (short)0, c, false, false);
}

// ---- hardware transcendental helpers (single TRANS32 ops, no libm fixups) ----
__device__ __forceinline__ float hw_exp2(float x) { return __builtin_amdgcn_exp2f(x); }
__device__ __forceinline__ float hw_log2(float x) { return __builtin_amdgcn_logf(x); }
__device__ __forceinline__ float hw_rcp(float x)  { return __builtin_amdgcn_rcpf(x); }

__device__ __forceinline__ float hw_tanh(float x) {
#if __has_builtin(__builtin_amdgcn_tanhf)
  return __builtin_amdgcn_tanhf(x);          // v_tanh_f32 (CDNA5 TRANS op)
#else
  const float e = hw_exp2(x * (2.0f * LOG2E_));
  return 1.0f - 2.0f * hw_rcp(e + 1.0f);     // (e^2x-1)/(e^2x+1), saturates to +-1
#endif
}

__device__ __forceinline__ float hw_sigmoid(float x) {
  return hw_rcp(1.0f + hw_exp2(-LOG2E_ * x));
}

__device__ __forceinline__ float hw_log(float x) {
  return hw_log2(x) * LN2_;
}

__global__ void __launch_bounds__(32 * NW)
toeplitz_lie_attn(const float* __restrict__ qg,
                  const float* __restrict__ kg,
                  const float* __restrict__ vg,
                  const float* __restrict__ lgg, const float* __restrict__ ltg,
                  const float* __restrict__ lgd, const float* __restrict__ ltd,
                  float* __restrict__ out)
{
  __shared__ float lds_att[NW][16][17];   // per-wave D->A relayout slice
  __shared__ float lds_v[NW][16][65];     // per-wave Vacc partials (pad kills conflicts)
  __shared__ float lds_ent[NW][16];       // per-wave entropy partials

  const int tid  = threadIdx.x;
  const int w    = tid >> 5;             // wave id 0..3
  const int lane = tid & 31;             // 0..31 (wave32)
  const int bid  = blockIdx.x;           // 0..1023
  const int rt   = bid & (TT - 1);       // row tile
  const int h    = (bid >> 6) & (H_ - 1);
  const int b    = bid >> 9;
  const int r0   = rt * 16;

  // learned scalar parameters (once per thread; exact libm here is cheap)
  const float gain_gate = fminf(fmaxf(expf(lgg[0]), 1e-3f), 20.0f);
  const float tau_gate  = fminf(fmaxf(expf(ltg[0]), 1e-3f), 10.0f);
  const float gain_dir  = fminf(fmaxf(expf(lgd[0]), 1e-3f), 20.0f);
  const float tau_dir   = fminf(fmaxf(expf(ltd[0]), 1e-3f), 10.0f);
  const float inv_tau_gate = 1.0f / tau_gate;
  const float inv_tau_dir  = 1.0f / tau_dir;
  const float gd_over_td   = gain_dir * inv_tau_dir;
  const float scale        = 0.125f;        // 1/sqrt(64)

  const size_t bhE  = (size_t)b * L_ * RS + (size_t)h * E_;
  const float* qptr = qg + bhE;
  const float* kptr = kg + bhE;
  const float* vptr = vg + bhE;

  const size_t V_SIZE   = (size_t)B_ * L_ * H_ * E_;    // 1048576
  const size_t ATT_SIZE = (size_t)B_ * H_ * L_ * L_;    // 16777216
  float* vout = out + bhE;                                        // (B,L,H,E)
  float* attb = out + V_SIZE + ((size_t)(b * H_ + h) * L_) * L_;  // (B,H,L,S)
  float* entb = out + V_SIZE + ATT_SIZE + (size_t)(b * H_ + h) * L_;

  // WMMA f32 16x16x4 lane mapping:
  //  A (16x4): lanes 0-15 -> M=lane hold K={0,1}; lanes 16-31 -> M=lane-16 hold K={2,3}
  //  B (4x16): lanes 0-15 -> N=lane hold K={0,1}; lanes 16-31 -> N=lane-16 hold K={2,3}
  //  C/D (16x16): VGPR m, lanes 0-15 -> (M=m, N=lane); lanes 16-31 -> (M=m+8, N=lane-16)
  const int mrow = lane & 15;
  const int koff = (lane >> 4) * 2;
  const int mhi  = (lane >= 16) ? 8 : 0;

  // Persistent A-layout tiles: Q rows [r0,r0+16) and K rows [r0,r0+16), full E=64.
  v2f Aq[16], Ak[16];
#pragma unroll
  for (int t = 0; t < 16; ++t) {
    const int e = 4 * t + koff;                      // even -> 8B aligned
    Aq[t] = *(const v2f*)(qptr + (size_t)(r0 + mrow) * RS + e);
    Ak[t] = *(const v2f*)(kptr + (size_t)(r0 + mrow) * RS + e);
  }

  v8f Vacc[4];
#pragma unroll
  for (int et = 0; et < 4; ++et) Vacc[et] = {};
  v8f ent = {};

  // Wave w handles column tiles j = w, w+NW, w+2*NW, ...
  for (int it = 0; it < TT / NW; ++it) {
    const int j  = w + NW * it;
    const int c0 = j * 16;

    // B-layout tiles for this column block: K rows and Q rows [c0, c0+16)
    v2f Bk[16], Bq[16];
#pragma unroll
    for (int t = 0; t < 16; ++t) {
      const int e = 4 * t + koff;
      Bk[t] = *(const v2f*)(kptr + (size_t)(c0 + mrow) * RS + e);
      Bq[t] = *(const v2f*)(qptr + (size_t)(c0 + mrow) * RS + e);
    }

    // Issue the value-tile loads NOW: their latency is covered by the 32
    // score WMMAs + the transcendental block below.
    v2f Vb[16];
#pragma unroll
    for (int et = 0; et < 4; ++et) {
#pragma unroll
      for (int t = 0; t < 4; ++t) {
        const int s = c0 + 4 * t + koff;
        const float* vr = vptr + (size_t)s * RS + 16 * et + mrow;
        v2f x; x.x = vr[0]; x.y = vr[RS];   // B[k][n], B[k+1][n]
        Vb[et * 4 + t] = x;
      }
    }

    // acc1 = Q_i K_j^T  (scores tile);  acc2 = K_i Q_j^T  (transposed scores tile,
    // same register layout -> symmetric/antisymmetric split is elementwise)
    v8f acc1 = {}, acc2 = {};
#pragma unroll
    for (int t = 0; t < 16; ++t) {
      acc1 = wmma4(Aq[t], Bk[t], acc1);
      acc2 = wmma4(Ak[t], Bq[t], acc2);
    }

    // gating / att / entropy, elementwise on the 8 accumulator rows per lane
#pragma unroll
    for (int m = 0; m < 8; ++m) {
      const float s1   = scale * acc1[m];        // scores[row][col]
      const float s2   = scale * acc2[m];        // scores[col][row]
      const float ssym = 0.5f * (s1 + s2);
      const float aant = 0.5f * (s1 - s2);

      const float gate = hw_sigmoid(gain_gate * hw_tanh(ssym * inv_tau_gate));
      const float dirp = hw_sigmoid(gain_dir  * hw_tanh(aant * inv_tau_dir));
      const int grow = r0 + m + mhi;
      const int gcol = c0 + mrow;
      const float Mm = (grow == gcol) ? 0.0f : (gate * dirp);  // zero diagonal

      float a = hw_tanh(aant * gd_over_td);
      a = fmaxf(a, 0.0f) * Mm;

      ent[m] += -a * hw_log(fmaxf(a, 1e-8f));    // a==0 contributes exactly 0

      attb[(size_t)grow * L_ + gcol] = a;        // att output, written once
      lds_att[w][m + mhi][mrow] = a;             // per-wave staging slice
    }
    // Same-wave DS ops are hardware in-order; we only need to stop the
    // compiler from reordering the LDS load above the stores.
    asm volatile("" ::: "memory");

    // att tile D-layout -> A-layout (K dim = key index s)
    v2f Aatt[4];
#pragma unroll
    for (int t = 0; t < 4; ++t) {
      const int kk = 4 * t + koff;
      v2f a2; a2.x = lds_att[w][mrow][kk]; a2.y = lds_att[w][mrow][kk + 1];
      Aatt[t] = a2;
    }
    asm volatile("" ::: "memory");

    // Vacc += att_tile (16x16) * value rows [c0,c0+16) (16x64)
#pragma unroll
    for (int et = 0; et < 4; ++et) {
#pragma unroll
      for (int t = 0; t < 4; ++t) {
        Vacc[et] = wmma4(Aatt[t], Vb[et * 4 + t], Vacc[et]);
      }
    }
  }

  // ---- per-wave entropy reduce across each 16-lane half (columns) ----
#pragma unroll
  for (int mask = 1; mask < 16; mask <<= 1) {
#pragma unroll
    for (int m = 0; m < 8; ++m)
      ent[m] += __shfl_xor(ent[m], mask, 32);
  }
  if (lane == 0 || lane == 16) {
#pragma unroll
    for (int m = 0; m < 8; ++m) lds_ent[w][mhi + m] = ent[m];
  }

  // ---- dump per-wave V partials ----
#pragma unroll
  for (int et = 0; et < 4; ++et) {
#pragma unroll
    for (int m = 0; m < 8; ++m) {
      lds_v[w][m + mhi][16 * et + mrow] = Vacc[et][m];
    }
  }
  __syncthreads();

  // ---- cooperative combine + store (128 threads over 16x64 outputs) ----
  {
    const int row   = tid >> 3;          // 0..15
    const int cbase = (tid & 7) * 8;     // 0,8,...,56
#pragma unroll
    for (int c = 0; c < 8; ++c) {
      const int col = cbase + c;
      float s = lds_v[0][row][col] + lds_v[1][row][col]
              + lds_v[2][row][col] + lds_v[3][row][col];
      vout[(size_t)(r0 + row) * RS + col] = s;
    }
  }
  if (tid < 16) {
    const float e = lds_ent[0][tid] + lds_ent[1][tid]
                  + lds_ent[2][tid] + lds_ent[3][tid];
    entb[r0 + tid] = e;
  }
}

extern "C" void kernel_launch(void* const* d_in, const int* in_sizes, int n_in,
                              void* d_out, int out_size, void* d_ws, size_t ws_size,
                              hipStream_t stream) {
  (void)in_sizes; (void)n_in; (void)out_size; (void)d_ws; (void)ws_size;
  const float* q   = (const float*)d_in[0];
  const float* k   = (const float*)d_in[1];
  const float* v   = (const float*)d_in[2];
  // d_in[3..6]: mask_miss_k, mask_miss_q, pos, causal_mask — unused by reference body
  const float* lgg = (const float*)d_in[7];
  const float* ltg = (const float*)d_in[8];
  const float* lgd = (const float*)d_in[9];
  const float* ltd = (const float*)d_in[10];

  dim3 grid(B_ * H_ * (L_ / 16));   // 1024 blocks, 4 wave32 each => 4096 waves
  toeplitz_lie_attn<<<grid, 32 * NW, 0, stream>>>(q, k, v, lgg, ltg, lgd, ltd,
                                                  (float*)d_out);
}